// AdaptiveRouter_83897891160582
// MI455X (gfx1250) — compile-verified
//
#include <hip/hip_runtime.h>
#include <hip/hip_bf16.h>
#include <math.h>

// ---------------------------------------------------------------------------
// AdaptiveRouter for MI455X (gfx1250, wave32)
//   hidden[32768,1024] f32 ; router_w[64,1024] ; w1[256,1024] ; b1[256]
//   w2[1,256] ; b2[1]
// Outputs (concat, f32): logits[N,64] | selected[N,4] | weights[N,4] | var | ent
// Single fused pass over `hidden` (HBM-dominant, 128MB) computing BOTH GEMMs
// with full-precision V_WMMA_F32_16X16X4_F32 + fused complexity head.
// A tile and B matrix are stored k-pair interleaved so every WMMA fragment is
// one 64-bit load straight into an even-aligned VGPR pair (no reg shuffles).
// ---------------------------------------------------------------------------

typedef float v2f __attribute__((ext_vector_type(2)));
typedef float v8f __attribute__((ext_vector_type(8)));

#define N_TOKENS   32768
#define HIDDEN     1024
#define NUM_EXP    64
#define H4         256
#define NCOLS      320            // 64 router cols + 256 w1 cols
#define MAX_K      4

#define M_TILE     64             // tokens per workgroup
#define KC         64             // K chunk staged in LDS
#define ASTR       (M_TILE * 2 + 2)   // 130: k-pair row stride (words)
#define WG_THREADS 640            // 20 waves: one 16-col stripe each
#define GEMM_GRID  (N_TOKENS / M_TILE)       // 512
#define EPI_BLK    256
#define EPI_GRID   (N_TOKENS / EPI_BLK)      // 128

// output offsets (floats)
#define OFF_SEL  ((size_t)N_TOKENS * NUM_EXP)              // 2097152
#define OFF_W    (OFF_SEL + (size_t)N_TOKENS * MAX_K)      // 2228224
#define OFF_VAR  (OFF_W + (size_t)N_TOKENS * MAX_K)        // 2359296
#define OFF_ENT  (OFF_VAR + 1)

// workspace offsets (floats)
#define WS_WT    0                                   // 1024*320 (pair-interleaved)
#define WS_CPLX  (HIDDEN * NCOLS)                    // + N_TOKENS
#define WS_PART  (WS_CPLX + N_TOKENS)                // + EPI_GRID*65

// ---------------------------------------------------------------------------
// Kernel 1: pair-interleaved weights:
//   Wt2[((k>>1)*NCOLS + c)*2 + (k&1)] = (c<64 ? router_w[c][k] : w1[c-64][k])
// Flat output index IS the loop index -> perfectly coalesced stores, and the
// GEMM's B fragment (k, k+1) @ column c becomes one global_load_b64.
// ---------------------------------------------------------------------------
__global__ __launch_bounds__(256)
void prep_wt_kernel(const float* __restrict__ router_w,
                    const float* __restrict__ w1,
                    float* __restrict__ Wt2) {
    int i = blockIdx.x * 256 + threadIdx.x;       // over 1024*320
    if (i >= HIDDEN * NCOLS) return;
    int b0 = i & 1;
    int r  = i >> 1;
    int c  = r % NCOLS;
    int k  = (r / NCOLS) * 2 + b0;
    float v = (c < NUM_EXP) ? router_w[(size_t)c * HIDDEN + k]
                            : w1[(size_t)(c - NUM_EXP) * HIDDEN + k];
    Wt2[i] = v;                                   // coalesced store
}

// ---------------------------------------------------------------------------
// Kernel 2: fused dual-GEMM + complexity head.
// 20 waves; wave w owns columns [16w, 16w+16) and 4 M-subtiles of 16 tokens.
// A (16x4 f32) frag: VGPR j, lane l -> m = l&15, k = k0 + j + 2*(l>>4)
// B (4x16 f32) frag: VGPR j, lane l -> n = l&15, k = k0 + j + 2*(l>>4)
// C/D (16x16 f32):   VGPR v, lane l -> n = l&15, m = v + 8*(l>>4)
// ---------------------------------------------------------------------------
__global__ __launch_bounds__(WG_THREADS)
void fused_gemm_kernel(const float* __restrict__ hidden,
                       const float* __restrict__ Wt2,
                       const float* __restrict__ b1,
                       const float* __restrict__ w2,
                       const float* __restrict__ b2,
                       float* __restrict__ out_logits,
                       float* __restrict__ complexity) {
    // k-pair interleaved A tile: element (k,m) at (k>>1)*ASTR + m*2 + (k&1)
    __shared__ float As2[(KC / 2) * ASTR];  // 4160 words = 16.6 KB
    __shared__ float tokAcc[M_TILE];        // per-token relu(h)·w2 partials

    const int tid   = threadIdx.x;
    const int wave  = tid >> 5;
    const int lane  = tid & 31;
    const int half  = lane >> 4;            // 0: lanes 0-15, 1: lanes 16-31
    const int l16   = lane & 15;
    const int tok0  = blockIdx.x * M_TILE;
    const int nbase = wave * 16;            // column stripe base (0..304)

    if (tid < M_TILE) tokAcc[tid] = 0.0f;

    v8f acc[4] = {};                        // 4 M-subtiles x 16x16 f32

    for (int kc = 0; kc < HIDDEN; kc += KC) {
        __syncthreads();
        // stage hidden[tok0..tok0+63][kc..kc+63] pair-interleaved into LDS.
        // Lanes carry consecutive k for fixed m: coalesced global reads,
        // conflict-free LDS writes (banks 2q+r distinct over 32 k values).
        for (int i = tid; i < M_TILE * KC; i += WG_THREADS) {
            int m = i >> 6;                 // i / KC
            int k = i & (KC - 1);
            As2[(k >> 1) * ASTR + m * 2 + (k & 1)] =
                hidden[(size_t)(tok0 + m) * HIDDEN + kc + k];
        }
        __syncthreads();

        for (int kk = 0; kk < KC; kk += 4) {
            const int kp = (kk >> 1) + half;        // k-pair index (in-chunk)

            // B fragment: one coalesced 8-byte load per lane
            v2f b = *(const v2f*)&Wt2[
                (size_t)(((kc + kk) >> 1) + half) * (NCOLS * 2) +
                (nbase + l16) * 2];

            // A fragments: one ds_load_b64 each, lands as an aligned pair
            v2f a[4];
#pragma unroll
            for (int s = 0; s < 4; ++s)
                a[s] = *(const v2f*)&As2[kp * ASTR + (16 * s + l16) * 2];

#pragma unroll
            for (int s = 0; s < 4; ++s) {
                acc[s] = __builtin_amdgcn_wmma_f32_16x16x4_f32(
                    /*neg_a=*/false, a[s], /*neg_b=*/false, b,
                    /*c_mod=*/(short)0, acc[s],
                    /*reuse_a=*/false, /*reuse_b=*/false);
            }
        }
    }

    if (nbase < NUM_EXP) {
        // router waves: store logits tiles (lane-coalesced 16-float runs)
#pragma unroll
        for (int s = 0; s < 4; ++s)
#pragma unroll
            for (int v = 0; v < 8; ++v) {
                int m = tok0 + 16 * s + v + 8 * half;
                out_logits[(size_t)m * NUM_EXP + nbase + l16] = acc[s][v];
            }
    } else {
        // w1 waves: bias + ReLU + dot with w2 column weight -> token partial
        const int   c1  = nbase - NUM_EXP + l16;   // 0..255
        const float w2v = w2[c1];
        const float b1v = b1[c1];
#pragma unroll
        for (int s = 0; s < 4; ++s)
#pragma unroll
            for (int v = 0; v < 8; ++v) {
                float h = acc[s][v] + b1v;
                h = h > 0.0f ? h : 0.0f;
                atomicAdd(&tokAcc[16 * s + v + 8 * half], h * w2v);
            }
    }
    __syncthreads();
    if (tid < M_TILE) {
        float x = tokAcc[tid] + b2[0];
        complexity[tok0 + tid] = 1.0f / (1.0f + expf(-x));
    }
}

// ---------------------------------------------------------------------------
// Kernel 3: per-token softmax stats + adaptive top-4 routing.
// One thread per token; per-block partials (no global atomics).
// ---------------------------------------------------------------------------
__global__ __launch_bounds__(EPI_BLK)
void epilogue_kernel(const float* __restrict__ logits,
                     const float* __restrict__ complexity,
                     float* __restrict__ out_sel,
                     float* __restrict__ out_w,
                     float* __restrict__ partials) {
    __shared__ float eSum[NUM_EXP];
    __shared__ float red[EPI_BLK];
    const int tid = threadIdx.x;
    if (tid < NUM_EXP) eSum[tid] = 0.0f;
    __syncthreads();

    const int tok = blockIdx.x * EPI_BLK + tid;
    const float* row = &logits[(size_t)tok * NUM_EXP];

    // full softmax (max-subtracted) + entropy + expert-load accumulation
    float mx = -INFINITY;
    for (int e = 0; e < NUM_EXP; ++e) mx = fmaxf(mx, row[e]);
    float sum = 0.0f;
    for (int e = 0; e < NUM_EXP; ++e) sum += expf(row[e] - mx);
    const float inv = 1.0f / sum;
    float ent = 0.0f;
    for (int e = 0; e < NUM_EXP; ++e) {
        float p = expf(row[e] - mx) * inv;
        ent -= p * logf(p + 1e-8f);
        atomicAdd(&eSum[e], p);            // block-local LDS reduction
    }

    // exact top-4, first-occurrence on ties (matches lax.top_k)
    float tv[MAX_K];
    int   ti[MAX_K];
    unsigned long long used = 0ull;
    for (int j = 0; j < MAX_K; ++j) {
        float best = -INFINITY;
        int   bi   = 0;
        for (int e = 0; e < NUM_EXP; ++e) {
            float v = row[e];
            if (!((used >> e) & 1ull) && v > best) { best = v; bi = e; }
        }
        tv[j] = best; ti[j] = bi;
        used |= 1ull << bi;
    }

    const int kAdapt = (complexity[tok] > 0.5f) ? MAX_K : 1;
    float wsum = 0.0f;
    float ew[MAX_K];
    for (int j = 0; j < MAX_K; ++j) {
        float e = (j < kAdapt) ? expf(tv[j] - tv[0]) : 0.0f;
        ew[j] = e; wsum += e;
    }
    const float winv = 1.0f / wsum;
    for (int j = 0; j < MAX_K; ++j) {
        out_w  [(size_t)tok * MAX_K + j] = (j < kAdapt) ? ew[j] * winv : 0.0f;
        out_sel[(size_t)tok * MAX_K + j] = (j < kAdapt) ? (float)ti[j] : 0.0f;
    }

    // entropy tree-reduce
    red[tid] = ent;
    __syncthreads();
    for (int s = EPI_BLK / 2; s > 0; s >>= 1) {
        if (tid < s) red[tid] += red[tid + s];
        __syncthreads();
    }
    if (tid < NUM_EXP) partials[(size_t)blockIdx.x * 65 + tid] = eSum[tid];
    if (tid == 0)      partials[(size_t)blockIdx.x * 65 + 64]  = red[0];
}

// ---------------------------------------------------------------------------
// Kernel 4: final reductions -> load_variance (ddof=1), mean entropy.
// ---------------------------------------------------------------------------
__global__ __launch_bounds__(128)
void finalize_kernel(const float* __restrict__ partials,
                     float* __restrict__ out) {
    __shared__ float loads[NUM_EXP];
    __shared__ float entM[1];
    const int tid = threadIdx.x;
    if (tid < NUM_EXP) {
        float s = 0.0f;
        for (int b = 0; b < EPI_GRID; ++b) s += partials[(size_t)b * 65 + tid];
        loads[tid] = s / (float)N_TOKENS;          // expert_load[e]
    }
    if (tid == NUM_EXP) {
        float s = 0.0f;
        for (int b = 0; b < EPI_GRID; ++b) s += partials[(size_t)b * 65 + 64];
        entM[0] = s / (float)N_TOKENS;
    }
    __syncthreads();
    if (tid == 0) {
        float mean = 0.0f;
        for (int e = 0; e < NUM_EXP; ++e) mean += loads[e];
        mean /= (float)NUM_EXP;
        float var = 0.0f;
        for (int e = 0; e < NUM_EXP; ++e) {
            float d = loads[e] - mean;
            var += d * d;
        }
        var /= (float)(NUM_EXP - 1);               // unbiased
        out[OFF_VAR] = var;
        out[OFF_ENT] = entM[0];
    }
}

// ---------------------------------------------------------------------------
extern "C" void kernel_launch(void* const* d_in, const int* in_sizes, int n_in,
                              void* d_out, int out_size, void* d_ws, size_t ws_size,
                              hipStream_t stream) {
    const float* hidden   = (const float*)d_in[0];
    const float* router_w = (const float*)d_in[1];
    const float* w1       = (const float*)d_in[2];
    const float* b1       = (const float*)d_in[3];
    const float* w2       = (const float*)d_in[4];
    const float* b2       = (const float*)d_in[5];
    float* out = (float*)d_out;
    float* ws  = (float*)d_ws;

    float* Wt2      = ws + WS_WT;
    float* cplx     = ws + WS_CPLX;
    float* partials = ws + WS_PART;

    prep_wt_kernel<<<(HIDDEN * NCOLS + 255) / 256, 256, 0, stream>>>(
        router_w, w1, Wt2);

    fused_gemm_kernel<<<GEMM_GRID, WG_THREADS, 0, stream>>>(
        hidden, Wt2, b1, w2, b2, out /*logits @ 0*/, cplx);

    epilogue_kernel<<<EPI_GRID, EPI_BLK, 0, stream>>>(
        out, cplx, out + OFF_SEL, out + OFF_W, partials);

    finalize_kernel<<<1, 128, 0, stream>>>(partials, out);
}